// LengthRegulator_83923660964128
// MI455X (gfx1250) — compile-verified
//
#include <hip/hip_runtime.h>
#include <hip/hip_bf16.h>
#include <stdint.h>

// LengthRegulator for MI455X (gfx1250, wave32).
// Pure bandwidth op: ~168 MB traffic -> ~7.2 us roofline @ 23.3 TB/s.
// x (33.5 MB) stays L2-resident; output stream uses non-temporal stores.

typedef float v4f __attribute__((ext_vector_type(4)));
typedef int   v4i __attribute__((ext_vector_type(4)));
typedef int   g4i __attribute__((vector_size(16)));   // GCC-style, matches builtin

typedef __attribute__((address_space(1))) g4i* g4i_global_ptr;
typedef __attribute__((address_space(3))) g4i* g4i_lds_ptr;

constexpr int B_      = 16;
constexpr int T_      = 1024;
constexpr int D_      = 512;
constexpr int MAXLEN_ = 4096;
constexpr int ROWS_   = 64;        // output rows handled per expand block
constexpr int D4_     = D_ / 4;    // 128 float4 slots per row

// ---------------------------------------------------------------------------
// Kernel 1: per-batch inclusive cumsum of durations (Hillis-Steele in LDS).
// ---------------------------------------------------------------------------
__global__ void __launch_bounds__(1024)
lr_scan_kernel(const int* __restrict__ dur, int* __restrict__ csum) {
  __shared__ int bufA[T_];
  __shared__ int bufB[T_];
  const int b = blockIdx.x;
  const int t = threadIdx.x;

  int v = dur[b * T_ + t];
  bufA[t] = (v < 0) ? 0 : v;
  __syncthreads();

  int* in  = bufA;
  int* out = bufB;
  for (int off = 1; off < T_; off <<= 1) {
    out[t] = (t >= off) ? (in[t] + in[t - off]) : in[t];
    __syncthreads();
    int* tmp = in; in = out; out = tmp;
  }
  csum[b * T_ + t] = in[t];
}

// ---------------------------------------------------------------------------
// Kernel 2: expand. Each block: one batch, 64 consecutive output positions.
//  - stage 4KB csum into LDS via async global->LDS (CDNA5 path) if available
//  - upper_bound binary search per position (== searchsorted side='right')
//  - 128-bit coalesced copy; NT stores for the streaming output.
// ---------------------------------------------------------------------------
__global__ void __launch_bounds__(256)
lr_expand_kernel(const float* __restrict__ x, const int* __restrict__ csum,
                 float* __restrict__ out) {
  __shared__ alignas(16) int sh_csum[T_];
  __shared__ int sh_src[ROWS_];

  const int tid            = threadIdx.x;
  const int blocksPerBatch = MAXLEN_ / ROWS_;   // 64
  const int b              = blockIdx.x / blocksPerBatch;
  const int rowBase        = (blockIdx.x % blocksPerBatch) * ROWS_;
  const int* cs            = csum + b * T_;

  // ---- stage csum[b][0..1023] into LDS (16 B per thread) ----
#if defined(__HIP_DEVICE_COMPILE__) &&                                        \
    __has_builtin(__builtin_amdgcn_global_load_async_to_lds_b128) &&          \
    __has_builtin(__builtin_amdgcn_s_wait_asynccnt)
  {
    __builtin_amdgcn_global_load_async_to_lds_b128(
        (g4i_global_ptr)(uintptr_t)(cs + tid * 4),
        (g4i_lds_ptr)(unsigned)(uintptr_t)&sh_csum[tid * 4],
        0, 0);
    __builtin_amdgcn_s_wait_asynccnt(0);
  }
#else
  ((v4i*)sh_csum)[tid] = ((const v4i*)cs)[tid];
#endif
  __syncthreads();

  const int total = sh_csum[T_ - 1];

  // ---- searchsorted(csum, pos, side='right'), clamped to T-1 ----
  if (tid < ROWS_) {
    const int pos = rowBase + tid;
    int lo = 0, hi = T_;
    while (lo < hi) {
      const int mid = (lo + hi) >> 1;
      if (sh_csum[mid] <= pos) lo = mid + 1; else hi = mid;
    }
    if (lo > T_ - 1) lo = T_ - 1;
    sh_src[tid] = (pos < total) ? lo : -1;   // -1 => zero row (mask)
  }
  __syncthreads();

  // ---- bulk copy: 2 rows per pass, 128 float4 lanes per row ----
  const int lane = tid & (D4_ - 1);   // float4 slot within row
  const int rsub = tid >> 7;          // 0 or 1
  const v4f* __restrict__ x4 = (const v4f*)x;
  v4f* __restrict__ o4       = (v4f*)out;
  const size_t xBatch = (size_t)b * T_ * D4_;
  const size_t oBase  = ((size_t)b * MAXLEN_ + (size_t)rowBase) * D4_;

#pragma unroll 4
  for (int p = 0; p < ROWS_ / 2; ++p) {
    const int r   = (p << 1) + rsub;
    const int src = sh_src[r];
    v4f v = 0;
    if (src >= 0) v = x4[xBatch + (size_t)src * D4_ + lane];
    __builtin_nontemporal_store(v, &o4[oBase + (size_t)r * D4_ + lane]);
  }
}

// ---------------------------------------------------------------------------
extern "C" void kernel_launch(void* const* d_in, const int* in_sizes, int n_in,
                              void* d_out, int out_size, void* d_ws, size_t ws_size,
                              hipStream_t stream) {
  const float* x   = (const float*)d_in[0];
  const int*   dur = (const int*)d_in[1];   // durations (values 0..7)
  // d_in[2] = max_len scalar == 4096 (compile-time constant here)

  int*   csum = (int*)d_ws;                 // 16*1024*4 = 64 KB scratch
  float* out  = (float*)d_out;              // (16, 4096, 512) f32

  lr_scan_kernel<<<dim3(B_), dim3(T_), 0, stream>>>(dur, csum);
  lr_expand_kernel<<<dim3(B_ * (MAXLEN_ / ROWS_)), dim3(256), 0, stream>>>(x, csum, out);

  (void)in_sizes; (void)n_in; (void)out_size; (void)ws_size;
}